// GraphEncoder_9405978378507
// MI455X (gfx1250) — compile-verified
//
#include <hip/hip_runtime.h>
#include <math.h>

// ---------------- problem constants (match reference) ----------------
#define NN     50000
#define EE     800000
#define ETOT   (EE + NN)      // edges + self loops
#define GG     512
#define FIN    300
#define HH     4
#define CC     64
#define HCC    256
#define NHID   300
#define NHIDP  304            // padded to /16
#define NOUT   512
#define NEG_SLOPE 0.2f
#define KCHUNK 64

typedef __attribute__((ext_vector_type(2))) float v2f;
typedef __attribute__((ext_vector_type(8))) float v8f;

static __host__ int cdiv(int a, int b) { return (a + b - 1) / b; }

// ------------------------- device helpers ---------------------------
__device__ __forceinline__ void edge_sd(const int* __restrict__ eidx, int e,
                                        int& s, int& d) {
  if (e < EE) { s = eidx[e]; d = eidx[EE + e]; }
  else        { s = e - EE;  d = e - EE; }
}

__device__ __forceinline__ void atomicMaxF(float* addr, float val) {
  if (val >= 0.0f) atomicMax((int*)addr, __float_as_int(val));
  else             atomicMin((unsigned int*)addr, (unsigned int)__float_as_int(val));
}

__device__ __forceinline__ float leaky(float v) {
  return v > 0.0f ? v : NEG_SLOPE * v;
}

// ---------------------------------------------------------------------
// GEMM (node features): C[M,256] = A[M,K] @ B[K,256]   (f32 WMMA 16x16x4)
// Block = 256 threads = 8 waves. Block owns a 16-row stripe; A-tile staged
// in LDS once and reused by all 8 waves (each wave -> 32 output columns).
// ---------------------------------------------------------------------
__global__ __launch_bounds__(256)
void gemm_node_kernel(const float* __restrict__ A, const float* __restrict__ B,
                      float* __restrict__ C, int M, int K) {
  __shared__ float As[16][KCHUNK + 4];
  const int row0  = blockIdx.x * 16;
  const int tid   = threadIdx.x;
  const int wave  = tid >> 5;
  const int lane  = tid & 31;
  const int m16   = lane & 15;
  const int khalf = (lane >> 4) * 2;      // lanes 0-15 -> K0/K1, 16-31 -> K2/K3
  const int col0  = wave * 32;

  v8f acc0 = {0.f,0.f,0.f,0.f,0.f,0.f,0.f,0.f};
  v8f acc1 = {0.f,0.f,0.f,0.f,0.f,0.f,0.f,0.f};

  for (int k0 = 0; k0 < K; k0 += KCHUNK) {
    const int kw = (K - k0 < KCHUNK) ? (K - k0) : KCHUNK;
    // stage A tile [16 x KCHUNK] (zero-padded) into LDS, coalesced
    #pragma unroll
    for (int i = 0; i < (16 * KCHUNK) / 256; ++i) {
      int linear = tid + i * 256;
      int mm = linear >> 6;          // KCHUNK == 64
      int kk = linear & (KCHUNK - 1);
      float v = 0.0f;
      if (kk < kw) v = A[(size_t)(row0 + mm) * K + (k0 + kk)];
      As[mm][kk] = v;
    }
    __syncthreads();
    // prefetch next A chunk into cache (global_prefetch_b8)
    if (k0 + KCHUNK < K)
      __builtin_prefetch(&A[(size_t)(row0 + (tid >> 4)) * K + k0 + KCHUNK +
                            ((tid & 15) << 2)], 0, 1);
    for (int kk = 0; kk < kw; kk += 4) {
      v2f a;
      a.x = As[m16][kk + khalf];
      a.y = As[m16][kk + khalf + 1];
      const float* Br0 = &B[(size_t)(k0 + kk + khalf) * HCC];
      const float* Br1 = Br0 + HCC;
      v2f b0, b1;
      b0.x = Br0[col0 + m16];       b0.y = Br1[col0 + m16];
      b1.x = Br0[col0 + 16 + m16];  b1.y = Br1[col0 + 16 + m16];
      acc0 = __builtin_amdgcn_wmma_f32_16x16x4_f32(false, a, false, b0,
                                                   (short)0, acc0, false, false);
      acc1 = __builtin_amdgcn_wmma_f32_16x16x4_f32(false, a, false, b1,
                                                   (short)0, acc1, false, false);
    }
    __syncthreads();
  }
  const int rofs = (lane >> 4) * 8;   // C/D layout: VGPR v -> row v (lo) / v+8 (hi)
  #pragma unroll
  for (int v = 0; v < 8; ++v) {
    C[(size_t)(row0 + v + rofs) * HCC + col0 + m16]      = acc0[v];
    C[(size_t)(row0 + v + rofs) * HCC + col0 + 16 + m16] = acc1[v];
  }
}

// ---------------------------------------------------------------------
// Generic small GEMM: one wave per 16x16 tile.  C = A@B (+bias)(+relu)
// M, Nc multiples of 16; K multiple of 4.
// ---------------------------------------------------------------------
__global__ __launch_bounds__(128)
void gemm_tile_kernel(const float* __restrict__ A, const float* __restrict__ B,
                      const float* __restrict__ bias, float* __restrict__ C,
                      int M, int K, int Nc, int do_relu) {
  const int wid    = blockIdx.x * (blockDim.x >> 5) + (threadIdx.x >> 5);
  const int ntiles = Nc >> 4;
  if (wid >= (M >> 4) * ntiles) return;      // wave-uniform exit (EXEC stays full)
  const int trow  = wid / ntiles;
  const int tcol  = wid % ntiles;
  const int lane  = threadIdx.x & 31;
  const int m16   = lane & 15;
  const int khalf = (lane >> 4) * 2;

  v8f acc = {0.f,0.f,0.f,0.f,0.f,0.f,0.f,0.f};
  for (int k = 0; k < K; k += 4) {
    v2f a, b;
    const float* Ar = &A[(size_t)(trow * 16 + m16) * K + k + khalf];
    a.x = Ar[0]; a.y = Ar[1];
    b.x = B[(size_t)(k + khalf)     * Nc + tcol * 16 + m16];
    b.y = B[(size_t)(k + khalf + 1) * Nc + tcol * 16 + m16];
    acc = __builtin_amdgcn_wmma_f32_16x16x4_f32(false, a, false, b,
                                                (short)0, acc, false, false);
  }
  const float bv   = bias ? bias[tcol * 16 + m16] : 0.0f;
  const int   rofs = (lane >> 4) * 8;
  #pragma unroll
  for (int v = 0; v < 8; ++v) {
    float o = acc[v] + bv;
    if (do_relu) o = fmaxf(o, 0.0f);
    C[(size_t)(trow * 16 + v + rofs) * Nc + tcol * 16 + m16] = o;
  }
}

// --------------------- attention score reduction ---------------------
// al_s[n,h] = sum_c h[n,h,c]*a_s[h,c] ; al_d analogous.
__global__ __launch_bounds__(256)
void att_kernel(const float* __restrict__ h, const float* __restrict__ a_s,
                const float* __restrict__ a_d, float* __restrict__ als,
                float* __restrict__ ald) {
  int idx = blockIdx.x * blockDim.x + threadIdx.x;
  if (idx >= NN * HH) return;
  int n = idx >> 2, hd = idx & 3;
  const float* hp = h + (size_t)n * HCC + hd * CC;
  const float* sp = a_s + hd * CC;
  const float* dp = a_d + hd * CC;
  float ss = 0.f, dd = 0.f;
  #pragma unroll 8
  for (int c = 0; c < CC; ++c) { float hv = hp[c]; ss += hv * sp[c]; dd += hv * dp[c]; }
  als[idx] = ss; ald[idx] = dd;
}

// out <- bias (broadcast), m <- -inf, s <- 0
__global__ __launch_bounds__(256)
void init_layer_kernel(float* __restrict__ out, const float* __restrict__ bias,
                       float* __restrict__ m, float* __restrict__ s) {
  int idx = blockIdx.x * blockDim.x + threadIdx.x;
  if (idx < NN * HCC) out[idx] = bias[idx & (HCC - 1)];
  if (idx < NN * HH) { m[idx] = __int_as_float(0xff800000); s[idx] = 0.0f; }
}

// pass 1: segment max of leaky(al_s[src]+al_d[dst]) into m[dst]
__global__ __launch_bounds__(256)
void edge_max_kernel(const int* __restrict__ eidx, const float* __restrict__ als,
                     const float* __restrict__ ald, float* __restrict__ m) {
  int idx = blockIdx.x * blockDim.x + threadIdx.x;
  if (idx >= ETOT * HH) return;
  int e = idx >> 2, hd = idx & 3, s, d;
  edge_sd(eidx, e, s, d);
  float ev = leaky(als[s * HH + hd] + ald[d * HH + hd]);
  atomicMaxF(&m[d * HH + hd], ev);
}

// pass 2: ex = exp(e - m[dst]); store ex; segment-sum into s[dst]
__global__ __launch_bounds__(256)
void edge_exp_kernel(const int* __restrict__ eidx, const float* __restrict__ als,
                     const float* __restrict__ ald, const float* __restrict__ m,
                     float* __restrict__ sbuf, float* __restrict__ exb) {
  int idx = blockIdx.x * blockDim.x + threadIdx.x;
  if (idx >= ETOT * HH) return;
  int e = idx >> 2, hd = idx & 3, s, d;
  edge_sd(eidx, e, s, d);
  float ev = leaky(als[s * HH + hd] + ald[d * HH + hd]);
  float ex = __expf(ev - m[d * HH + hd]);
  exb[idx] = ex;
  atomicAdd(&sbuf[d * HH + hd], ex);
}

// pass 3: one wave per edge, scatter alpha * h[src] into out[dst]
__global__ __launch_bounds__(256)
void aggregate_kernel(const int* __restrict__ eidx, const float* __restrict__ h,
                      const float* __restrict__ exb, const float* __restrict__ sbuf,
                      float* __restrict__ out) {
  int wid = blockIdx.x * 8 + (threadIdx.x >> 5);
  if (wid >= ETOT) return;
  int lane = threadIdx.x & 31, s, d;
  edge_sd(eidx, wid, s, d);
  float a0 = exb[wid * 4 + 0] / (sbuf[d * 4 + 0] + 1e-16f);
  float a1 = exb[wid * 4 + 1] / (sbuf[d * 4 + 1] + 1e-16f);
  float a2 = exb[wid * 4 + 2] / (sbuf[d * 4 + 2] + 1e-16f);
  float a3 = exb[wid * 4 + 3] / (sbuf[d * 4 + 3] + 1e-16f);
  const float* hs = h + (size_t)s * HCC;
  float* od = out + (size_t)d * HCC;
  #pragma unroll
  for (int k = 0; k < 8; ++k) {
    int c = k * 32 + lane;
    float alpha = (c < 64) ? a0 : (c < 128) ? a1 : (c < 192) ? a2 : a3;
    atomicAdd(&od[c], alpha * hs[c]);
  }
}

__global__ __launch_bounds__(256)
void relu_kernel(float* __restrict__ p, int n) {
  int idx = blockIdx.x * blockDim.x + threadIdx.x;
  if (idx < n) p[idx] = fmaxf(p[idx], 0.0f);
}

// ------------------------------ pooling ------------------------------
__global__ __launch_bounds__(256)
void pool_zero_kernel(float* __restrict__ sums, float* __restrict__ cnt) {
  int idx = blockIdx.x * blockDim.x + threadIdx.x;
  if (idx < GG * HCC) sums[idx] = 0.0f;
  if (idx < GG) cnt[idx] = 0.0f;
}

__global__ __launch_bounds__(256)
void pool_accum_kernel(const float* __restrict__ h, const int* __restrict__ batch,
                       float* __restrict__ sums) {
  int idx = blockIdx.x * blockDim.x + threadIdx.x;
  if (idx >= NN * HCC) return;
  int n = idx >> 8, c = idx & (HCC - 1);
  atomicAdd(&sums[batch[n] * HCC + c], h[idx]);
}

__global__ __launch_bounds__(256)
void pool_count_kernel(const int* __restrict__ batch, float* __restrict__ cnt) {
  int idx = blockIdx.x * blockDim.x + threadIdx.x;
  if (idx < NN) atomicAdd(&cnt[batch[idx]], 1.0f);
}

__global__ __launch_bounds__(256)
void pool_div_kernel(const float* __restrict__ sums, const float* __restrict__ cnt,
                     float* __restrict__ pooled) {
  int idx = blockIdx.x * blockDim.x + threadIdx.x;
  if (idx >= GG * HCC) return;
  pooled[idx] = sums[idx] / fmaxf(cnt[idx >> 8], 1.0f);
}

// ------------------------ MLP weight padding -------------------------
__global__ __launch_bounds__(256)
void pad_w1_kernel(const float* __restrict__ w, float* __restrict__ wp) {
  int idx = blockIdx.x * blockDim.x + threadIdx.x;  // [256, 304]
  if (idx >= HCC * NHIDP) return;
  int r = idx / NHIDP, c = idx % NHIDP;
  wp[idx] = (c < NHID) ? w[r * NHID + c] : 0.0f;
}
__global__ __launch_bounds__(256)
void pad_b1_kernel(const float* __restrict__ b, float* __restrict__ bp) {
  int idx = blockIdx.x * blockDim.x + threadIdx.x;
  if (idx < NHIDP) bp[idx] = (idx < NHID) ? b[idx] : 0.0f;
}
__global__ __launch_bounds__(256)
void pad_w2_kernel(const float* __restrict__ w, float* __restrict__ wp) {
  int idx = blockIdx.x * blockDim.x + threadIdx.x;  // [304, 512]
  if (idx >= NHIDP * NOUT) return;
  int r = idx / NOUT, c = idx % NOUT;
  wp[idx] = (r < NHID) ? w[r * NOUT + c] : 0.0f;
}

// ------------------------------ launcher -----------------------------
extern "C" void kernel_launch(void* const* d_in, const int* in_sizes, int n_in,
                              void* d_out, int out_size, void* d_ws, size_t ws_size,
                              hipStream_t stream) {
  (void)in_sizes; (void)n_in; (void)out_size; (void)ws_size;
  const float* x    = (const float*)d_in[0];
  const int*   eidx = (const int*)  d_in[1];
  const int*   batch= (const int*)  d_in[2];
  const float* W1   = (const float*)d_in[3];
  const float* a1s  = (const float*)d_in[4];
  const float* a1d  = (const float*)d_in[5];
  const float* b1   = (const float*)d_in[6];
  const float* W2   = (const float*)d_in[7];
  const float* a2s  = (const float*)d_in[8];
  const float* a2d  = (const float*)d_in[9];
  const float* b2   = (const float*)d_in[10];
  const float* W3   = (const float*)d_in[11];
  const float* a3s  = (const float*)d_in[12];
  const float* a3d  = (const float*)d_in[13];
  const float* b3   = (const float*)d_in[14];
  const float* Wm1  = (const float*)d_in[15];
  const float* bm1  = (const float*)d_in[16];
  const float* Wm2  = (const float*)d_in[17];
  const float* bm2  = (const float*)d_in[18];

  float* ws = (float*)d_ws;
  size_t off = 0;
  auto alloc = [&](size_t n) { float* p = ws + off; off += (n + 63) & ~(size_t)63; return p; };
  float* hA     = alloc((size_t)NN * HCC);
  float* hB     = alloc((size_t)NN * HCC);
  float* als    = alloc((size_t)NN * HH);
  float* ald    = alloc((size_t)NN * HH);
  float* mb     = alloc((size_t)NN * HH);
  float* sb     = alloc((size_t)NN * HH);
  float* exb    = alloc((size_t)ETOT * HH);
  float* sums   = alloc((size_t)GG * HCC);
  float* cnt    = alloc((size_t)GG);
  float* pooled = alloc((size_t)GG * HCC);
  float* W1p    = alloc((size_t)HCC * NHIDP);
  float* b1p    = alloc((size_t)NHIDP);
  float* W2p    = alloc((size_t)NHIDP * NOUT);
  float* z      = alloc((size_t)GG * NHIDP);

  auto run_layer = [&](const float* in, int K, const float* W, const float* as_,
                       const float* ad_, const float* bias, float* hlin,
                       float* out, bool do_relu) {
    gemm_node_kernel<<<NN / 16, 256, 0, stream>>>(in, W, hlin, NN, K);
    att_kernel<<<cdiv(NN * HH, 256), 256, 0, stream>>>(hlin, as_, ad_, als, ald);
    init_layer_kernel<<<cdiv(NN * HCC, 256), 256, 0, stream>>>(out, bias, mb, sb);
    edge_max_kernel<<<cdiv(ETOT * HH, 256), 256, 0, stream>>>(eidx, als, ald, mb);
    edge_exp_kernel<<<cdiv(ETOT * HH, 256), 256, 0, stream>>>(eidx, als, ald, mb, sb, exb);
    aggregate_kernel<<<cdiv(ETOT, 8), 256, 0, stream>>>(eidx, hlin, exb, sb, out);
    if (do_relu) relu_kernel<<<cdiv(NN * HCC, 256), 256, 0, stream>>>(out, NN * HCC);
  };

  run_layer(x,  FIN, W1, a1s, a1d, b1, hA, hB, true);
  run_layer(hB, HCC, W2, a2s, a2d, b2, hA, hB, true);
  run_layer(hB, HCC, W3, a3s, a3d, b3, hA, hB, false);

  // global mean pool
  pool_zero_kernel <<<cdiv(GG * HCC, 256), 256, 0, stream>>>(sums, cnt);
  pool_accum_kernel<<<cdiv(NN * HCC, 256), 256, 0, stream>>>(hB, batch, sums);
  pool_count_kernel<<<cdiv(NN, 256), 256, 0, stream>>>(batch, cnt);
  pool_div_kernel  <<<cdiv(GG * HCC, 256), 256, 0, stream>>>(sums, cnt, pooled);

  // MLP head (pad 300 -> 304 so all WMMA tiles are full)
  pad_w1_kernel<<<cdiv(HCC * NHIDP, 256), 256, 0, stream>>>(Wm1, W1p);
  pad_b1_kernel<<<cdiv(NHIDP, 256), 256, 0, stream>>>(bm1, b1p);
  pad_w2_kernel<<<cdiv(NHIDP * NOUT, 256), 256, 0, stream>>>(Wm2, W2p);
  {
    int tiles1 = (GG / 16) * (NHIDP / 16);   // 32*19
    gemm_tile_kernel<<<cdiv(tiles1, 4), 128, 0, stream>>>(pooled, W1p, b1p, z,
                                                          GG, HCC, NHIDP, 1);
    int tiles2 = (GG / 16) * (NOUT / 16);    // 32*32
    gemm_tile_kernel<<<cdiv(tiles2, 4), 128, 0, stream>>>(z, W2p, bm2,
                                                          (float*)d_out,
                                                          GG, NHIDP, NOUT, 0);
  }
}